// LinearTransformerModel_55628416418518
// MI455X (gfx1250) — compile-verified
//
#include <hip/hip_runtime.h>

typedef float v2f __attribute__((ext_vector_type(2)));
typedef float v8f __attribute__((ext_vector_type(8)));

#define NROWS 16384
#define DDIM  256

// ws layout (in floats):
//   part : 256 tiles * 16 slots * 256 elems = 1,048,576 floats (4 MB)
//   M    : 256*256 = 65,536 floats
//   w    : 256 floats
#define WS_PART 0
#define WS_M    (256 * 16 * 256)
#define WS_W    (WS_M + 256 * 256)

// ---------------------------------------------------------------------------
// Kernel 1: partial Gram matrix  M = X^T X  via V_WMMA_F32_16X16X4_F32.
// Grid: (64, 2) blocks of 256 threads. Each block owns a 2x2 group of 16x16
// output tiles; each of its 8 waves handles a 1024-row K-chunk (kc selects
// which half of K). Partials go to distinct slots -> deterministic reduction.
// ---------------------------------------------------------------------------
__global__ __launch_bounds__(256)
void xtx_wmma_kernel(const float* __restrict__ X, float* __restrict__ part) {
  const int lane = threadIdx.x & 31;
  const int wid  = threadIdx.x >> 5;
  const int i2   = blockIdx.x >> 3;   // 0..7  (pair of row-tiles)
  const int j2   = blockIdx.x & 7;    // 0..7  (pair of col-tiles)
  const int kc   = blockIdx.y;        // 0..1
  const int slot = kc * 8 + wid;      // 0..15
  const int kBase = slot * 1024;

  // ISA 16x4 fp32 A layout: lanes 0-15 carry K=0(v0)/K=1(v1) for M=lane,
  // lanes 16-31 carry K=2/K=3 for M=lane-16. B (4x16) mirrors with N on lanes.
  const int col    = lane & 15;
  const int rowOff = (lane < 16) ? 0 : 2;

  const float* pA = X + (size_t)(kBase + rowOff) * DDIM + (i2 * 32 + col);
  const float* pB = X + (size_t)(kBase + rowOff) * DDIM + (j2 * 32 + col);

  v8f c00 = {}, c01 = {}, c10 = {}, c11 = {};

  for (int k = 0; k < 1024; k += 4) {
    v2f a0, a1, b0, b1;
    a0.x = pA[0];    a0.y = pA[DDIM];
    a1.x = pA[16];   a1.y = pA[DDIM + 16];
    b0.x = pB[0];    b0.y = pB[DDIM];
    b1.x = pB[16];   b1.y = pB[DDIM + 16];
    c00 = __builtin_amdgcn_wmma_f32_16x16x4_f32(false, a0, false, b0, (short)0, c00, false, false);
    c01 = __builtin_amdgcn_wmma_f32_16x16x4_f32(false, a0, false, b1, (short)0, c01, false, false);
    c10 = __builtin_amdgcn_wmma_f32_16x16x4_f32(false, a1, false, b0, (short)0, c10, false, false);
    c11 = __builtin_amdgcn_wmma_f32_16x16x4_f32(false, a1, false, b1, (short)0, c11, false, false);
    pA += 4 * DDIM;
    pB += 4 * DDIM;
  }

  // C/D layout: VGPR r -> M = r (lanes 0-15) or 8+r (lanes 16-31), N = lane&15
  const int m0 = (lane < 16) ? 0 : 8;
  const int n  = lane & 15;
#pragma unroll
  for (int a = 0; a < 2; ++a) {
#pragma unroll
    for (int b = 0; b < 2; ++b) {
      const int t = (2 * i2 + a) * 16 + (2 * j2 + b);
      float* P = part + ((size_t)t * 16 + slot) * 256;
      v8f cc = (a == 0) ? (b == 0 ? c00 : c01) : (b == 0 ? c10 : c11);
#pragma unroll
      for (int r = 0; r < 8; ++r)
        P[(m0 + r) * 16 + n] = cc[r];
    }
  }
}

// ---------------------------------------------------------------------------
// Kernel 2: deterministic fixed-order reduction of the 16 K-slot partials
// into the full 256x256 Gram matrix M.
// ---------------------------------------------------------------------------
__global__ __launch_bounds__(256)
void reduce_kernel(const float* __restrict__ part, float* __restrict__ M) {
  const int t  = blockIdx.x;      // tile 0..255
  const int e  = threadIdx.x;     // element 0..255 within tile
  const int ti = (t >> 4) * 16;
  const int tj = (t & 15) * 16;
  const int m  = e >> 4;
  const int n  = e & 15;
  float s = 0.f;
#pragma unroll
  for (int slot = 0; slot < 16; ++slot)
    s += part[((size_t)t * 16 + slot) * 256 + e];
  M[(size_t)(ti + m) * DDIM + (tj + n)] = s;
}

// ---------------------------------------------------------------------------
// Kernel 3: w = W * (M * b)   (two 256x256 matvecs, single workgroup)
// ---------------------------------------------------------------------------
__global__ __launch_bounds__(256)
void head_kernel(const float* __restrict__ M, const float* __restrict__ W,
                 const float* __restrict__ bvec, float* __restrict__ wout) {
  __shared__ float sb[DDIM];
  __shared__ float su[DDIM];
  const int i = threadIdx.x;
  sb[i] = bvec[i];
  __syncthreads();
  float acc = 0.f;
  for (int j = 0; j < DDIM; ++j) acc += M[(size_t)i * DDIM + j] * sb[j];
  su[i] = acc;
  __syncthreads();
  float acc2 = 0.f;
  for (int j = 0; j < DDIM; ++j) acc2 += W[(size_t)i * DDIM + j] * su[j];
  wout[i] = acc2;
}

// ---------------------------------------------------------------------------
// Kernel 4: g = X * w  (wave-per-row dot product, wave32 shuffle reduction)
// ---------------------------------------------------------------------------
__global__ __launch_bounds__(256)
void xw_kernel(const float* __restrict__ X, const float* __restrict__ w,
               float* __restrict__ g) {
  __shared__ float sw[DDIM];
  sw[threadIdx.x] = w[threadIdx.x];
  __syncthreads();
  const int lane = threadIdx.x & 31;
  const int wid  = threadIdx.x >> 5;
  const int row  = blockIdx.x * 8 + wid;
  const float* xr = X + (size_t)row * DDIM;
  float acc = 0.f;
#pragma unroll
  for (int t = 0; t < DDIM; t += 32)
    acc += xr[t + lane] * sw[t + lane];
#pragma unroll
  for (int off = 16; off > 0; off >>= 1)
    acc += __shfl_xor(acc, off, 32);
  if (lane == 0) g[row] = acc;
}

// ---------------------------------------------------------------------------
extern "C" void kernel_launch(void* const* d_in, const int* in_sizes, int n_in,
                              void* d_out, int out_size, void* d_ws, size_t ws_size,
                              hipStream_t stream) {
  const float* X = (const float*)d_in[0];   // (16384, 256)
  const float* W = (const float*)d_in[1];   // (256, 256)
  const float* b = (const float*)d_in[2];   // (256,)
  float* g   = (float*)d_out;               // (16384,)
  float* wsf = (float*)d_ws;

  float* part = wsf + WS_PART;
  float* M    = wsf + WS_M;
  float* wv   = wsf + WS_W;

  xtx_wmma_kernel<<<dim3(64, 2), 256, 0, stream>>>(X, part);
  reduce_kernel<<<256, 256, 0, stream>>>(part, M);
  head_kernel<<<1, 256, 0, stream>>>(M, W, b, wv);
  xw_kernel<<<NROWS / 8, 256, 0, stream>>>(X, wv, g);
}